// Temporal_GAT_Transformer_27977416966568
// MI455X (gfx1250) — compile-verified
//
#include <hip/hip_runtime.h>
#include <hip/hip_bf16.h>
#include <math.h>

// ---------------------------------------------------------------------------
// Temporal GAT Transformer, fp32 end-to-end, CDNA5 (gfx1250) wave32 + WMMA.
// Heavy GEMMs use V_WMMA_F32_16X16X4_F32 (full fp32, no conversion error).
// Encoder layer fully fused per block, 2 nodes per block so every weight
// B-fragment load feeds two WMMA ops (halves L2 weight traffic).
// GAT phases use a device-built CSR so segment softmax/aggregation are
// wave reductions, not atomics.  Workspace requirement: ~226 MB.
// ---------------------------------------------------------------------------

#define N_NODES 8192
#define T_STEPS 16
#define IN_F    8
#define DMODEL  128
#define E_EDGES (N_NODES * 16 + N_NODES)   // 139264
#define NT      (N_NODES * T_STEPS)        // 131072
#define NEG_SLOPE 0.2f
#define LDX 132   // padded LDS row (kills bank conflicts on A-fragment reads)

typedef float v2f __attribute__((ext_vector_type(2)));
typedef float v8f __attribute__((ext_vector_type(8)));

__device__ __forceinline__ v8f wmma4(v2f a, v2f b, v8f c) {
  // D = A(16x4) * B(4x16) + C, fp32
  return __builtin_amdgcn_wmma_f32_16x16x4_f32(false, a, false, b, (short)0, c,
                                               false, false);
}
__device__ __forceinline__ v8f zero8() {
  v8f z;
#pragma unroll
  for (int i = 0; i < 8; i++) z[i] = 0.f;
  return z;
}
__device__ __forceinline__ float wred_sum(float v) {
#pragma unroll
  for (int m = 16; m >= 1; m >>= 1) v += __shfl_xor(v, m, 32);
  return v;
}
__device__ __forceinline__ float wred_max(float v) {
#pragma unroll
  for (int m = 16; m >= 1; m >>= 1) v = fmaxf(v, __shfl_xor(v, m, 32));
  return v;
}
// reductions within each 16-lane half (matches WMMA D-fragment row groups)
__device__ __forceinline__ float hred_sum(float v) {
#pragma unroll
  for (int m = 8; m >= 1; m >>= 1) v += __shfl_xor(v, m, 32);
  return v;
}
__device__ __forceinline__ float hred_max(float v) {
#pragma unroll
  for (int m = 8; m >= 1; m >>= 1) v = fmaxf(v, __shfl_xor(v, m, 32));
  return v;
}
__device__ __forceinline__ float leakyr(float a) {
  return a >= 0.f ? a : NEG_SLOPE * a;
}

// ---------------------------------------------------------------------------
// CSR build (edges grouped by dst; dst is identical across all timesteps)
// ---------------------------------------------------------------------------
__global__ void k_init(int* cnt, float* node) {
  int i = blockIdx.x * blockDim.x + threadIdx.x;
  if (i < N_NODES) { cnt[i] = 0; node[i] = 0.f; }
}
__global__ void k_count(const int* ei, int* cnt) {
  int e = blockIdx.x * blockDim.x + threadIdx.x;
  if (e < E_EDGES) atomicAdd(&cnt[ei[E_EDGES + e]], 1);
}
__global__ void k_scan(const int* cnt, int* off, int* cursor) {
  __shared__ int part[256];
  int th = threadIdx.x, base = th * 32, s = 0;
  for (int i = 0; i < 32; i++) s += cnt[base + i];
  part[th] = s;
  __syncthreads();
  if (th == 0) {
    int run = 0;
    for (int i = 0; i < 256; i++) { int t = part[i]; part[i] = run; run += t; }
    off[N_NODES] = run;
  }
  __syncthreads();
  int acc = part[th];
  for (int i = 0; i < 32; i++) {
    off[base + i] = acc; cursor[base + i] = acc; acc += cnt[base + i];
  }
}
__global__ void k_scatter(const int* ei, int* cursor, int* eid) {
  int e = blockIdx.x * blockDim.x + threadIdx.x;
  if (e < E_EDGES) {
    int d = ei[E_EDGES + e];
    int pos = atomicAdd(&cursor[d], 1);
    eid[pos] = e;
  }
}

// ---------------------------------------------------------------------------
// GAT1:  H[NT,256] = feature[NT,8] @ W1[8,256]   (WMMA, K=8)
// ---------------------------------------------------------------------------
__global__ __launch_bounds__(256) void k_gat1_h(const float* __restrict__ F,
                                                const float* __restrict__ W1,
                                                float* __restrict__ H) {
  int tid = threadIdx.x, w = tid >> 5, l = tid & 31, r = l & 15, half = l >> 4;
  int id = blockIdx.x * 8 + w;         // (NT/16)*(256/16) = 131072 wave tiles
  int mt = id >> 4, nt = id & 15, n0 = nt * 16;
  v8f acc = zero8();
#pragma unroll
  for (int k0 = 0; k0 < 8; k0 += 4) {
    int kb = k0 + 2 * half;
    v2f a; a.x = F[(size_t)(mt * 16 + r) * 8 + kb];
           a.y = F[(size_t)(mt * 16 + r) * 8 + kb + 1];
    v2f b; b.x = W1[kb * 256 + n0 + r];
           b.y = W1[(kb + 1) * 256 + n0 + r];
    acc = wmma4(a, b, acc);
  }
#pragma unroll
  for (int j = 0; j < 8; j++) {
    int m = j + 8 * half;
    H[(size_t)(mt * 16 + m) * 256 + n0 + r] = acc[j];
  }
}

// s_src[nt,head] = sum_c H[nt,head*128+c]*att_src[head,c]; same for dst
__global__ __launch_bounds__(256) void k_gat1_s(const float* __restrict__ H,
                                                const float* __restrict__ asrc,
                                                const float* __restrict__ adst,
                                                float* ssrc, float* sdst) {
  int tid = threadIdx.x, w = tid >> 5, l = tid & 31;
  int nt = blockIdx.x * 8 + w;
  const float* row = H + (size_t)nt * 256;
  float s0 = 0, s1 = 0, d0 = 0, d1 = 0;
#pragma unroll
  for (int i = 0; i < 8; i++) {
    int c = l + 32 * i;
    float v = row[c], as = asrc[c], ad = adst[c];
    if (c < 128) { s0 += v * as; d0 += v * ad; }
    else         { s1 += v * as; d1 += v * ad; }
  }
  s0 = wred_sum(s0); s1 = wred_sum(s1);
  d0 = wred_sum(d0); d1 = wred_sum(d1);
  if (l == 0) {
    ssrc[nt * 2] = s0; ssrc[nt * 2 + 1] = s1;
    sdst[nt * 2] = d0; sdst[nt * 2 + 1] = d1;
  }
}

// wave per (node,t): segment softmax over incoming edges + aggregation,
// head-mean + bias + positional encoding -> X[NT,128]
__global__ __launch_bounds__(256) void k_gat1_agg(
    const float* __restrict__ H, const float* __restrict__ ssrc,
    const float* __restrict__ sdst, const int* __restrict__ ei,
    const int* __restrict__ off, const int* __restrict__ eid,
    const float* __restrict__ bias1, float* __restrict__ X) {
  int tid = threadIdx.x, w = tid >> 5, l = tid & 31;
  int nt = blockIdx.x * 8 + w;
  int n = nt >> 4, t = nt & 15;
  int o0 = off[n], deg = off[n + 1] - o0;
  float sd0 = sdst[nt * 2], sd1 = sdst[nt * 2 + 1];
  float m0 = -1e30f, m1 = -1e30f;
  for (int j = l; j < deg; j += 32) {
    int e = eid[o0 + j]; int s = ei[e];
    int sb = (s * 16 + t) * 2;
    m0 = fmaxf(m0, leakyr(ssrc[sb] + sd0));
    m1 = fmaxf(m1, leakyr(ssrc[sb + 1] + sd1));
  }
  m0 = wred_max(m0); m1 = wred_max(m1);
  float z0 = 0, z1 = 0;
  float a0c[4] = {0, 0, 0, 0}, a1c[4] = {0, 0, 0, 0};
  for (int j = 0; j < deg; j++) {     // lanes cover channels, edges serial
    int e = eid[o0 + j]; int s = ei[e];
    int sb = (s * 16 + t) * 2;
    float e0 = expf(leakyr(ssrc[sb] + sd0) - m0);
    float e1 = expf(leakyr(ssrc[sb + 1] + sd1) - m1);
    z0 += e0; z1 += e1;
    const float* hr = H + (size_t)(s * 16 + t) * 256;
#pragma unroll
    for (int r2 = 0; r2 < 4; r2++) {
      a0c[r2] += e0 * hr[l + 32 * r2];
      a1c[r2] += e1 * hr[128 + l + 32 * r2];
    }
  }
  float iz0 = 1.f / z0, iz1 = 1.f / z1;
  const float LND = 9.210340371976184f / 128.f;  // ln(10000)/128
#pragma unroll
  for (int r2 = 0; r2 < 4; r2++) {
    int c = l + 32 * r2;
    float div = expf(-(float)(c & ~1) * LND);
    float pe = (c & 1) ? cosf((float)t * div) : sinf((float)t * div);
    X[(size_t)nt * 128 + c] =
        0.5f * (a0c[r2] * iz0 + a1c[r2] * iz1) + bias1[c] + pe;
  }
}

__global__ void k_copy_tg(const float* X, float* Tg) {
  int i = blockIdx.x * blockDim.x + threadIdx.x;
  if (i < N_NODES * 128) {
    int n = i >> 7, c = i & 127;
    Tg[i] = X[(size_t)(n * 16 + 15) * 128 + c];
  }
}

// ---------------------------------------------------------------------------
// Fused post-LN transformer encoder layer.
// 2 nodes per block (32 seq rows in LDS): each weight B-fragment load is
// reused by two WMMA A-tiles -> 2x arithmetic intensity on weight traffic.
// ---------------------------------------------------------------------------
struct EncP {
  const float *wq, *wk, *wv, *wo, *bq, *bk, *bv, *bo;
  const float *w1, *b1, *w2, *b2;
  const float *g1, *bb1, *g2, *bb2;
};

__global__ __launch_bounds__(256) void k_enc_layer(float* __restrict__ Xg, EncP p) {
  __shared__ float Xl[32][LDX], Ql[32][LDX], Kl[32][LDX], Vl[32][LDX], Yl[32][LDX];
  __shared__ float Pl[16][16][17];   // [node*8+head][q][key]
  __shared__ float Hl[32][20];
  int tid = threadIdx.x, w = tid >> 5, l = tid & 31, r = l & 15, half = l >> 4;
  float* Xrow = Xg + (size_t)blockIdx.x * 4096;   // 2 nodes = 32 rows x 128
  for (int i = tid; i < 4096; i += 256) Xl[i >> 7][i & 127] = Xrow[i];
  __syncthreads();
  int n0 = w * 16;
  // ---- Q,K,V projections (wave w -> output cols [16w,16w+16), both nodes) ----
  {
    v8f aq[2], ak[2], av[2];
#pragma unroll
    for (int u = 0; u < 2; u++) { aq[u] = zero8(); ak[u] = zero8(); av[u] = zero8(); }
    for (int k0 = 0; k0 < 128; k0 += 4) {
      int kb = k0 + 2 * half;
      v2f bQ; bQ.x = p.wq[kb * 128 + n0 + r]; bQ.y = p.wq[(kb + 1) * 128 + n0 + r];
      v2f bK; bK.x = p.wk[kb * 128 + n0 + r]; bK.y = p.wk[(kb + 1) * 128 + n0 + r];
      v2f bV; bV.x = p.wv[kb * 128 + n0 + r]; bV.y = p.wv[(kb + 1) * 128 + n0 + r];
#pragma unroll
      for (int u = 0; u < 2; u++) {
        v2f a; a.x = Xl[u * 16 + r][kb]; a.y = Xl[u * 16 + r][kb + 1];
        aq[u] = wmma4(a, bQ, aq[u]);
        ak[u] = wmma4(a, bK, ak[u]);
        av[u] = wmma4(a, bV, av[u]);
      }
    }
    float bq_ = p.bq[n0 + r], bk_ = p.bk[n0 + r], bv_ = p.bv[n0 + r];
#pragma unroll
    for (int u = 0; u < 2; u++)
#pragma unroll
      for (int j = 0; j < 8; j++) {
        int m = u * 16 + j + 8 * half;
        Ql[m][n0 + r] = aq[u][j] + bq_;
        Kl[m][n0 + r] = ak[u][j] + bk_;
        Vl[m][n0 + r] = av[u][j] + bv_;
      }
  }
  __syncthreads();
  // ---- scores = Qh @ Kh^T / 4, softmax (wave w == head, both nodes) ----
  int ch0 = 16 * w;
#pragma unroll
  for (int u = 0; u < 2; u++) {
    int tt = u * 8 + w;
    v8f s8 = zero8();
#pragma unroll
    for (int kk = 0; kk < 16; kk += 4) {
      int kd = kk + 2 * half;
      v2f a; a.x = Ql[u * 16 + r][ch0 + kd]; a.y = Ql[u * 16 + r][ch0 + kd + 1];
      v2f b; b.x = Kl[u * 16 + r][ch0 + kd]; b.y = Kl[u * 16 + r][ch0 + kd + 1];
      s8 = wmma4(a, b, s8);
    }
#pragma unroll
    for (int j = 0; j < 8; j++) {           // softmax over keys (lane dim)
      float v = s8[j] * 0.25f;
      float mx = hred_max(v);
      float ev = expf(v - mx);
      float sm = hred_sum(ev);
      Pl[tt][j + 8 * half][r] = ev / sm;
    }
  }
  __syncthreads();
  // ---- O_head = P @ V_head (store into Ql cols of own head) ----
#pragma unroll
  for (int u = 0; u < 2; u++) {
    int tt = u * 8 + w;
    v8f o8 = zero8();
#pragma unroll
    for (int kk = 0; kk < 16; kk += 4) {
      int kd = kk + 2 * half;
      v2f a; a.x = Pl[tt][r][kd];            a.y = Pl[tt][r][kd + 1];
      v2f b; b.x = Vl[u * 16 + kd][ch0 + r]; b.y = Vl[u * 16 + kd + 1][ch0 + r];
      o8 = wmma4(a, b, o8);
    }
#pragma unroll
    for (int j = 0; j < 8; j++) Ql[u * 16 + j + 8 * half][ch0 + r] = o8[j];
  }
  __syncthreads();
  // ---- attn_out @ wo + bo + residual ----
  {
    v8f ao[2]; ao[0] = zero8(); ao[1] = zero8();
    for (int k0 = 0; k0 < 128; k0 += 4) {
      int kb = k0 + 2 * half;
      v2f b; b.x = p.wo[kb * 128 + n0 + r]; b.y = p.wo[(kb + 1) * 128 + n0 + r];
#pragma unroll
      for (int u = 0; u < 2; u++) {
        v2f a; a.x = Ql[u * 16 + r][kb]; a.y = Ql[u * 16 + r][kb + 1];
        ao[u] = wmma4(a, b, ao[u]);
      }
    }
    float bo_ = p.bo[n0 + r];
#pragma unroll
    for (int u = 0; u < 2; u++)
#pragma unroll
      for (int j = 0; j < 8; j++) {
        int m = u * 16 + j + 8 * half;
        Yl[m][n0 + r] = ao[u][j] + bo_ + Xl[m][n0 + r];
      }
  }
  __syncthreads();
  // ---- LN1 -> Xl (4 rows per wave) ----
  for (int rr = 0; rr < 4; rr++) {
    int row = 4 * w + rr;
    float vals[4], s = 0, ss = 0;
#pragma unroll
    for (int i = 0; i < 4; i++) {
      float v = Yl[row][l + 32 * i]; vals[i] = v; s += v; ss += v * v;
    }
    s = wred_sum(s); ss = wred_sum(ss);
    float mu = s * (1.f / 128.f);
    float rs = rsqrtf(ss * (1.f / 128.f) - mu * mu + 1e-5f);
#pragma unroll
    for (int i = 0; i < 4; i++) {
      int c = l + 32 * i;
      Xl[row][c] = (vals[i] - mu) * rs * p.g1[c] + p.bb1[c];
    }
  }
  __syncthreads();
  // ---- FFN hidden = relu(X @ w1 + b1), one M-tile per wave 0/1 ----
  if (w < 2) {
    int u = w;
    v8f hh = zero8();
    for (int k0 = 0; k0 < 128; k0 += 4) {
      int kb = k0 + 2 * half;
      v2f a; a.x = Xl[u * 16 + r][kb]; a.y = Xl[u * 16 + r][kb + 1];
      v2f b; b.x = p.w1[kb * 16 + r];  b.y = p.w1[(kb + 1) * 16 + r];
      hh = wmma4(a, b, hh);
    }
    float b1_ = p.b1[r];
#pragma unroll
    for (int j = 0; j < 8; j++)
      Hl[u * 16 + j + 8 * half][r] = fmaxf(hh[j] + b1_, 0.f);
  }
  __syncthreads();
  // ---- FFN out = hidden @ w2 + b2 + residual ----
  {
    v8f fo[2]; fo[0] = zero8(); fo[1] = zero8();
#pragma unroll
    for (int kk = 0; kk < 16; kk += 4) {
      int kd = kk + 2 * half;
      v2f b; b.x = p.w2[kd * 128 + n0 + r]; b.y = p.w2[(kd + 1) * 128 + n0 + r];
#pragma unroll
      for (int u = 0; u < 2; u++) {
        v2f a; a.x = Hl[u * 16 + r][kd]; a.y = Hl[u * 16 + r][kd + 1];
        fo[u] = wmma4(a, b, fo[u]);
      }
    }
    float b2_ = p.b2[n0 + r];
#pragma unroll
    for (int u = 0; u < 2; u++)
#pragma unroll
      for (int j = 0; j < 8; j++) {
        int m = u * 16 + j + 8 * half;
        Yl[m][n0 + r] = fo[u][j] + b2_ + Xl[m][n0 + r];
      }
  }
  __syncthreads();
  // ---- LN2 -> global (4 rows per wave) ----
  for (int rr = 0; rr < 4; rr++) {
    int row = 4 * w + rr;
    float vals[4], s = 0, ss = 0;
#pragma unroll
    for (int i = 0; i < 4; i++) {
      float v = Yl[row][l + 32 * i]; vals[i] = v; s += v; ss += v * v;
    }
    s = wred_sum(s); ss = wred_sum(ss);
    float mu = s * (1.f / 128.f);
    float rs = rsqrtf(ss * (1.f / 128.f) - mu * mu + 1e-5f);
#pragma unroll
    for (int i = 0; i < 4; i++) {
      int c = l + 32 * i;
      Xrow[row * 128 + c] = (vals[i] - mu) * rs * p.g2[c] + p.bb2[c];
    }
  }
}

// generic per-row layernorm (wave per row), in place
__global__ __launch_bounds__(256) void k_ln_rows(float* X, const float* g,
                                                 const float* b, int M) {
  int w = threadIdx.x >> 5, l = threadIdx.x & 31;
  int row = blockIdx.x * 8 + w;
  if (row >= M) return;
  float* Y = X + (size_t)row * 128;
  float vals[4], s = 0, ss = 0;
#pragma unroll
  for (int i = 0; i < 4; i++) {
    float v = Y[l + 32 * i]; vals[i] = v; s += v; ss += v * v;
  }
  s = wred_sum(s); ss = wred_sum(ss);
  float mu = s * (1.f / 128.f);
  float rs = rsqrtf(ss * (1.f / 128.f) - mu * mu + 1e-5f);
#pragma unroll
  for (int i = 0; i < 4; i++) {
    int c = l + 32 * i;
    Y[c] = (vals[i] - mu) * rs * g[c] + b[c];
  }
}

// ---------------------------------------------------------------------------
// Fused decoder layer: block per node. Self-attn with L=1 collapses exactly
// to (t@wv+bv)@wo+bo. Cross K/V projections use WMMA (the heavy part).
// ---------------------------------------------------------------------------
struct DecP {
  const float *swv, *swo, *sbv, *sbo;                 // self (q,k unused: L=1)
  const float *cwq, *cwk, *cwv, *cwo, *cbq, *cbk, *cbv, *cbo;
  const float *w1, *b1, *w2, *b2;
  const float *g1, *bb1, *g2, *bb2, *g3, *bb3;
};

__device__ __forceinline__ void dec_ln(const float* Y, const float* g,
                                       const float* b, float* dst, int w, int l) {
  if (w == 0) {
    float vals[4], s = 0, ss = 0;
#pragma unroll
    for (int i = 0; i < 4; i++) {
      float v = Y[l + 32 * i]; vals[i] = v; s += v; ss += v * v;
    }
    s = wred_sum(s); ss = wred_sum(ss);
    float mu = s * (1.f / 128.f);
    float rs = rsqrtf(ss * (1.f / 128.f) - mu * mu + 1e-5f);
#pragma unroll
    for (int i = 0; i < 4; i++) {
      int c = l + 32 * i;
      dst[c] = (vals[i] - mu) * rs * g[c] + b[c];
    }
  }
}

__global__ __launch_bounds__(256) void k_dec_layer(const float* __restrict__ Mem,
                                                   float* __restrict__ Tg, DecP p) {
  __shared__ float Ml[16][LDX], Kl[16][LDX], Vl[16][LDX];
  __shared__ float Tl[128], Aux[128], Yt[128], Qt[128], Ot[128], Ht[16];
  __shared__ float Pw[8][16];
  int tid = threadIdx.x, w = tid >> 5, l = tid & 31, r = l & 15, half = l >> 4;
  const float* Mr = Mem + (size_t)blockIdx.x * 2048;
  for (int i = tid; i < 2048; i += 256) Ml[i >> 7][i & 127] = Mr[i];
  if (tid < 128) Tl[tid] = Tg[(size_t)blockIdx.x * 128 + tid];
  __syncthreads();
  // ---- self-attn (exact, softmax of 1x1 == 1) ----
  if (tid < 128) {
    float a = 0;
    for (int k = 0; k < 128; k++) a += Tl[k] * p.swv[k * 128 + tid];
    Aux[tid] = a + p.sbv[tid];
  }
  __syncthreads();
  if (tid < 128) {
    float a = 0;
    for (int k = 0; k < 128; k++) a += Aux[k] * p.swo[k * 128 + tid];
    Yt[tid] = a + p.sbo[tid] + Tl[tid];
  }
  __syncthreads();
  dec_ln(Yt, p.g1, p.bb1, Tl, w, l);
  __syncthreads();
  // ---- cross-attn ----
  if (tid < 128) {
    float a = 0;
    for (int k = 0; k < 128; k++) a += Tl[k] * p.cwq[k * 128 + tid];
    Qt[tid] = a + p.cbq[tid];
  }
  {  // K,V = mem @ wk/wv + bias  (WMMA, wave w -> cols [16w,16w+16))
    int n0 = w * 16;
    v8f ak = zero8(), av = zero8();
    for (int k0 = 0; k0 < 128; k0 += 4) {
      int kb = k0 + 2 * half;
      v2f a;  a.x = Ml[r][kb]; a.y = Ml[r][kb + 1];
      v2f bK; bK.x = p.cwk[kb * 128 + n0 + r]; bK.y = p.cwk[(kb + 1) * 128 + n0 + r];
      v2f bV; bV.x = p.cwv[kb * 128 + n0 + r]; bV.y = p.cwv[(kb + 1) * 128 + n0 + r];
      ak = wmma4(a, bK, ak); av = wmma4(a, bV, av);
    }
    float bk_ = p.cbk[n0 + r], bv_ = p.cbv[n0 + r];
#pragma unroll
    for (int j = 0; j < 8; j++) {
      int m = j + 8 * half;
      Kl[m][n0 + r] = ak[j] + bk_;
      Vl[m][n0 + r] = av[j] + bv_;
    }
  }
  __syncthreads();
  {  // scores & softmax over 16 keys (wave w == head, lane%16 == key)
    float s = 0;
    for (int d = 0; d < 16; d++) s += Qt[16 * w + d] * Kl[r][16 * w + d];
    s *= 0.25f;
    float mx = hred_max(s);
    float ev = expf(s - mx);
    float sm = hred_sum(ev);
    if (half == 0) Pw[w][r] = ev / sm;
  }
  __syncthreads();
  if (half == 0) {  // o[16h+d] = sum_k p_k * V[k][16h+d]
    float o = 0;
    for (int k = 0; k < 16; k++) o += Pw[w][k] * Vl[k][16 * w + r];
    Ot[16 * w + r] = o;
  }
  __syncthreads();
  if (tid < 128) {
    float a = 0;
    for (int k = 0; k < 128; k++) a += Ot[k] * p.cwo[k * 128 + tid];
    Yt[tid] = a + p.cbo[tid] + Tl[tid];
  }
  __syncthreads();
  dec_ln(Yt, p.g2, p.bb2, Tl, w, l);
  __syncthreads();
  // ---- FFN ----
  if (tid < 16) {
    float a = 0;
    for (int k = 0; k < 128; k++) a += Tl[k] * p.w1[k * 16 + tid];
    Ht[tid] = fmaxf(a + p.b1[tid], 0.f);
  }
  __syncthreads();
  if (tid < 128) {
    float a = 0;
    for (int k = 0; k < 16; k++) a += Ht[k] * p.w2[k * 128 + tid];
    Yt[tid] = a + p.b2[tid] + Tl[tid];
  }
  __syncthreads();
  dec_ln(Yt, p.g3, p.bb3, Tl, w, l);
  __syncthreads();
  if (tid < 128) Tg[(size_t)blockIdx.x * 128 + tid] = Tl[tid];
}

// ---------------------------------------------------------------------------
// GAT2: s_src/s_dst = (Tg @ W2) * att  (WMMA with fused epilogue)
// ---------------------------------------------------------------------------
__global__ __launch_bounds__(256) void k_gat2(const float* __restrict__ Tg,
                                              const float* __restrict__ W2,
                                              const float* __restrict__ as2,
                                              const float* __restrict__ ad2,
                                              float* s2s, float* s2d) {
  int tid = threadIdx.x, w = tid >> 5, l = tid & 31, r = l & 15, half = l >> 4;
  int id = blockIdx.x * 8 + w;           // (8192/16)*(128/16) = 4096 tiles
  int mt = id >> 3, ntile = id & 7, n0 = ntile * 16;
  v8f acc = zero8();
  for (int k0 = 0; k0 < 128; k0 += 4) {
    int kb = k0 + 2 * half;
    v2f a; a.x = Tg[(size_t)(mt * 16 + r) * 128 + kb];
           a.y = Tg[(size_t)(mt * 16 + r) * 128 + kb + 1];
    v2f b; b.x = W2[kb * 128 + n0 + r];
           b.y = W2[(kb + 1) * 128 + n0 + r];
    acc = wmma4(a, b, acc);
  }
  int n = n0 + r;
  float as_ = as2[n], ad_ = ad2[n];
#pragma unroll
  for (int j = 0; j < 8; j++) {
    int row = mt * 16 + j + 8 * half;
    s2s[(size_t)row * 128 + n] = acc[j] * as_;
    s2d[(size_t)row * 128 + n] = acc[j] * ad_;
  }
}

// wave per dst node: per-head segment max and sum-of-exp
__global__ __launch_bounds__(256) void k_gat2_mz(const float* __restrict__ s2s,
                                                 const float* __restrict__ s2d,
                                                 const int* __restrict__ ei,
                                                 const int* __restrict__ off,
                                                 const int* __restrict__ eid,
                                                 float* m2, float* z2) {
  int tid = threadIdx.x, w = tid >> 5, l = tid & 31;
  int n = blockIdx.x * 8 + w;
  int o0 = off[n], deg = off[n + 1] - o0;
  float sd[4], mx[4], z[4];
#pragma unroll
  for (int rr = 0; rr < 4; rr++) {
    sd[rr] = s2d[(size_t)n * 128 + l + 32 * rr];
    mx[rr] = -1e30f; z[rr] = 0.f;
  }
  for (int j = 0; j < deg; j++) {
    int e = eid[o0 + j]; int s = ei[e];
    const float* sr = s2s + (size_t)s * 128;
#pragma unroll
    for (int rr = 0; rr < 4; rr++)
      mx[rr] = fmaxf(mx[rr], leakyr(sr[l + 32 * rr] + sd[rr]));
  }
  for (int j = 0; j < deg; j++) {
    int e = eid[o0 + j]; int s = ei[e];
    const float* sr = s2s + (size_t)s * 128;
#pragma unroll
    for (int rr = 0; rr < 4; rr++)
      z[rr] += expf(leakyr(sr[l + 32 * rr] + sd[rr]) - mx[rr]);
  }
#pragma unroll
  for (int rr = 0; rr < 4; rr++) {
    m2[(size_t)n * 128 + l + 32 * rr] = mx[rr];
    z2[(size_t)n * 128 + l + 32 * rr] = z[rr];
  }
}

// wave per edge: flow = sum_h alpha_h * conv_w[h] + conv_b ; write flow_cat
__global__ __launch_bounds__(256) void k_flow(const float* __restrict__ s2s,
                                              const float* __restrict__ s2d,
                                              const float* __restrict__ m2,
                                              const float* __restrict__ z2,
                                              const int* __restrict__ ei,
                                              const float* __restrict__ cw,
                                              const float* __restrict__ cb,
                                              float* __restrict__ out) {
  int tid = threadIdx.x, w = tid >> 5, l = tid & 31;
  int e = blockIdx.x * 8 + w;
  if (e >= E_EDGES) return;
  int s = ei[e], d = ei[E_EDGES + e];
  float acc = 0;
#pragma unroll
  for (int rr = 0; rr < 4; rr++) {
    int h = l + 32 * rr;
    float a = leakyr(s2s[(size_t)s * 128 + h] + s2d[(size_t)d * 128 + h]);
    float al = expf(a - m2[(size_t)d * 128 + h]) / z2[(size_t)d * 128 + h];
    acc += al * cw[h];
  }
  acc = wred_sum(acc);
  if (l == 0) {
    size_t base = (size_t)N_NODES + (size_t)e * 3;
    out[base] = (float)s;
    out[base + 1] = (float)d;
    out[base + 2] = acc + cb[0];
  }
}

__global__ void k_phys_edges(const float* __restrict__ out,
                             const float* __restrict__ F,
                             const int* __restrict__ ei, float* node) {
  int e = blockIdx.x * blockDim.x + threadIdx.x;
  if (e >= E_EDGES) return;
  int s = ei[e], d = ei[E_EDGES + e];
  if (s == d) return;
  float f = out[(size_t)N_NODES + (size_t)e * 3 + 2];
  float contrib = f * F[(size_t)(s * 16 + 15) * 8 + 0];  // conc[src]
  atomicAdd(&node[s], -contrib / F[(size_t)(s * 16 + 15) * 8 + 2]);
  atomicAdd(&node[d],  contrib / F[(size_t)(d * 16 + 15) * 8 + 2]);
}

__global__ void k_phys_final(const float* __restrict__ F,
                             const float* __restrict__ node, float* out) {
  int n = blockIdx.x * blockDim.x + threadIdx.x;
  if (n >= N_NODES) return;
  const float* fr = F + (size_t)(n * 16 + 15) * 8;
  float mask = (n == N_NODES - 1) ? 0.f : 1.f;
  out[n] = fr[0] + (node[n] + 0.005f * fr[1] / fr[2]) * mask;
}

// ---------------------------------------------------------------------------
// Host launcher
// ---------------------------------------------------------------------------
extern "C" void kernel_launch(void* const* d_in, const int* in_sizes, int n_in,
                              void* d_out, int out_size, void* d_ws,
                              size_t ws_size, hipStream_t stream) {
  // d_in layout: setup_inputs() dict insertion order, flattened recursively.
  const float* F   = (const float*)d_in[0];   // feature [N,T,8]
  const int*   EI  = (const int*)d_in[1];     // edge_index [2,E]
  const float* W1  = (const float*)d_in[2];   // gat1.W [8,256]
  const float* AS1 = (const float*)d_in[3];   // gat1.att_src [2,128]
  const float* AD1 = (const float*)d_in[4];   // gat1.att_dst [2,128]
  const float* B1g = (const float*)d_in[5];   // gat1.bias [128]
  const float* W2  = (const float*)d_in[6];   // gat2.W [128,128]
  const float* AS2 = (const float*)d_in[7];   // gat2.att_src [128,1]
  const float* AD2 = (const float*)d_in[8];   // gat2.att_dst [128,1]
  // d_in[9] = gat2.bias (unused: reference returns alpha pre-bias)
  const float* ENG = (const float*)d_in[94];  // enc_norm.g
  const float* ENB = (const float*)d_in[95];  // enc_norm.b
  const float* DNG = (const float*)d_in[96];  // dec_norm.g
  const float* DNB = (const float*)d_in[97];  // dec_norm.b
  const float* CW  = (const float*)d_in[98];  // conv_w [128]
  const float* CB  = (const float*)d_in[99];  // conv_b [1]

  float* ws = (float*)d_ws;
  size_t o = 0;
  float* X   = ws + o; o += (size_t)NT * 128;        // 16.8M
  float* H   = ws + o; o += (size_t)NT * 256;        // 33.6M
  float* SS1 = ws + o; o += (size_t)NT * 2;
  float* SD1 = ws + o; o += (size_t)NT * 2;
  float* Tg  = ws + o; o += (size_t)N_NODES * 128;
  float* S2S = ws + o; o += (size_t)N_NODES * 128;
  float* S2D = ws + o; o += (size_t)N_NODES * 128;
  float* M2  = ws + o; o += (size_t)N_NODES * 128;
  float* Z2  = ws + o; o += (size_t)N_NODES * 128;
  float* NODE = ws + o; o += N_NODES;
  int* ib  = (int*)(ws + o);
  int* CNT = ib;
  int* CUR = ib + N_NODES;
  int* OFF = CUR + N_NODES;                // N_NODES+1 ints
  int* EID = OFF + N_NODES + 1;            // E ints

  float* OUT = (float*)d_out;

  // CSR build (dst grouping, shared by GAT1 and GAT2)
  k_init<<<32, 256, 0, stream>>>(CNT, NODE);
  k_count<<<(E_EDGES + 255) / 256, 256, 0, stream>>>(EI, CNT);
  k_scan<<<1, 256, 0, stream>>>(CNT, OFF, CUR);
  k_scatter<<<(E_EDGES + 255) / 256, 256, 0, stream>>>(EI, CUR, EID);

  // GAT1 (all timesteps fused)
  k_gat1_h<<<16384, 256, 0, stream>>>(F, W1, H);
  k_gat1_s<<<NT / 8, 256, 0, stream>>>(H, AS1, AD1, SS1, SD1);
  k_gat1_agg<<<NT / 8, 256, 0, stream>>>(H, SS1, SD1, EI, OFF, EID, B1g, X);

  // decoder query t = x[:, -1] BEFORE encoder
  k_copy_tg<<<(N_NODES * 128 + 255) / 256, 256, 0, stream>>>(X, Tg);

  // encoder (in-place on X, 2 nodes per block)
  for (int lidx = 0; lidx < 2; lidx++) {
    int b = 10 + lidx * 16;  // wq,wk,wv,wo,bq,bk,bv,bo,w1,b1,w2,b2,g1,b1,g2,b2
    EncP p;
    p.wq = (const float*)d_in[b + 0];  p.wk = (const float*)d_in[b + 1];
    p.wv = (const float*)d_in[b + 2];  p.wo = (const float*)d_in[b + 3];
    p.bq = (const float*)d_in[b + 4];  p.bk = (const float*)d_in[b + 5];
    p.bv = (const float*)d_in[b + 6];  p.bo = (const float*)d_in[b + 7];
    p.w1 = (const float*)d_in[b + 8];  p.b1 = (const float*)d_in[b + 9];
    p.w2 = (const float*)d_in[b + 10]; p.b2 = (const float*)d_in[b + 11];
    p.g1 = (const float*)d_in[b + 12]; p.bb1 = (const float*)d_in[b + 13];
    p.g2 = (const float*)d_in[b + 14]; p.bb2 = (const float*)d_in[b + 15];
    k_enc_layer<<<N_NODES / 2, 256, 0, stream>>>(X, p);
  }
  k_ln_rows<<<NT / 8, 256, 0, stream>>>(X, ENG, ENB, NT);  // enc_norm

  // decoder (in-place on Tg, mem = X)
  for (int lidx = 0; lidx < 2; lidx++) {
    int b = 42 + lidx * 26;  // self(8), cross(8), w1,b1,w2,b2, ln1..ln3
    DecP p;
    p.swv = (const float*)d_in[b + 2];  p.swo = (const float*)d_in[b + 3];
    p.sbv = (const float*)d_in[b + 6];  p.sbo = (const float*)d_in[b + 7];
    p.cwq = (const float*)d_in[b + 8];  p.cwk = (const float*)d_in[b + 9];
    p.cwv = (const float*)d_in[b + 10]; p.cwo = (const float*)d_in[b + 11];
    p.cbq = (const float*)d_in[b + 12]; p.cbk = (const float*)d_in[b + 13];
    p.cbv = (const float*)d_in[b + 14]; p.cbo = (const float*)d_in[b + 15];
    p.w1 = (const float*)d_in[b + 16];  p.b1 = (const float*)d_in[b + 17];
    p.w2 = (const float*)d_in[b + 18];  p.b2 = (const float*)d_in[b + 19];
    p.g1 = (const float*)d_in[b + 20];  p.bb1 = (const float*)d_in[b + 21];
    p.g2 = (const float*)d_in[b + 22];  p.bb2 = (const float*)d_in[b + 23];
    p.g3 = (const float*)d_in[b + 24];  p.bb3 = (const float*)d_in[b + 25];
    k_dec_layer<<<N_NODES, 256, 0, stream>>>(X, Tg, p);
  }
  k_ln_rows<<<N_NODES / 8, 256, 0, stream>>>(Tg, DNG, DNB, N_NODES);  // dec_norm

  // GAT2 alpha -> flow, physics
  k_gat2<<<512, 256, 0, stream>>>(Tg, W2, AS2, AD2, S2S, S2D);
  k_gat2_mz<<<N_NODES / 8, 256, 0, stream>>>(S2S, S2D, EI, OFF, EID, M2, Z2);
  k_flow<<<(E_EDGES + 7) / 8, 256, 0, stream>>>(S2S, S2D, M2, Z2, EI, CW, CB, OUT);
  k_phys_edges<<<(E_EDGES + 255) / 256, 256, 0, stream>>>(OUT, F, EI, NODE);
  k_phys_final<<<32, 256, 0, stream>>>(F, NODE, OUT);
}